// QuestionConvNetwork_85727547228593
// MI455X (gfx1250) — compile-verified
//
#include <hip/hip_runtime.h>
#include <stdint.h>

#define N_NODES   100000
#define N_EDGES   3200000
#define D_FEAT    128
#define NUM_LAYERS 3
#define KEEP      0.9f

#define BLOCK     256
#define WAVE_SZ   32
#define WAVES     (BLOCK / WAVE_SZ)        // 8 waves per block
#define BATCH     64                       // edges per block-iteration
#define EPW       (BATCH / WAVES)          // 8 edges per wave == TDM gather cap

typedef uint32_t u32x4 __attribute__((ext_vector_type(4)));
typedef uint32_t u32x8 __attribute__((ext_vector_type(8)));

// ---------------------------------------------------------------------------
// CDNA5 async copy: global -> LDS, 4 bytes per active lane (ASYNCcnt tracked).
// ---------------------------------------------------------------------------
__device__ __forceinline__ void async_ld_dword_to_lds(uint32_t lds_byte_off,
                                                      const void* base,
                                                      uint32_t byte_off) {
  asm volatile("global_load_async_to_lds_b32 %0, %1, %2"
               :: "v"(lds_byte_off), "v"(byte_off), "s"(base)
               : "memory");
}

__device__ __forceinline__ void wait_asynccnt0() {
  asm volatile("s_wait_asynccnt 0x0" ::: "memory");
}

__device__ __forceinline__ uint32_t lds_off_of(const void* p) {
  // Flat LDS-aperture address = {aperture_hi, lds_byte_off} -> low 32 bits.
  return (uint32_t)(uintptr_t)p;
}

// Counter-based dropout hash -> uniform [0,1)
__device__ __forceinline__ float rng01(uint32_t layer, uint32_t idx) {
  uint32_t x = idx * 0x9E3779B9u + (layer + 1u) * 0x85EBCA6Bu + 0xC2B2AE35u;
  x ^= x >> 16; x *= 0x7FEB352Du;
  x ^= x >> 15; x *= 0x846CA68Bu;
  x ^= x >> 16;
  return (float)(x >> 8) * (1.0f / 16777216.0f);
}

// ---------------------------------------------------------------------------
// acc = prev = x ; h = 0
// ---------------------------------------------------------------------------
__global__ void __launch_bounds__(BLOCK)
gcn_init(const float* __restrict__ x, float* __restrict__ out,
         float* __restrict__ prev, float* __restrict__ hbuf) {
  const int n4 = N_NODES * D_FEAT / 4;
  int i = blockIdx.x * BLOCK + threadIdx.x;
  if (i >= n4) return;
  float4 v = reinterpret_cast<const float4*>(x)[i];
  reinterpret_cast<float4*>(out)[i]  = v;
  reinterpret_cast<float4*>(prev)[i] = v;
  reinterpret_cast<float4*>(hbuf)[i] = make_float4(0.f, 0.f, 0.f, 0.f);
}

// ---------------------------------------------------------------------------
// COO SpMM: h[r,:] += v * x[c,:]
//   - edge stream staged via ASYNCcnt global->LDS copies (double buffered)
//   - feature-row gather done by the Tensor Data Mover in gather mode
//     (8 x 32-bit row indices per D#), double buffered against the
//     FMA + global_atomic_add_f32 scatter of the previous batch.
// ---------------------------------------------------------------------------
__global__ void __launch_bounds__(BLOCK)
gcn_spmm(const int* __restrict__ rows, const int* __restrict__ cols,
         const float* __restrict__ vals, const float* __restrict__ x,
         float* __restrict__ h) {
  __shared__ int   s_cols[2][BATCH];
  __shared__ int   s_rows[2][BATCH];
  __shared__ float s_vals[2][BATCH];
  __shared__ float s_xtile[2][WAVES][EPW][D_FEAT];   // 64 KB gathered rows

  const int tid  = threadIdx.x;
  const int wave = tid >> 5;
  const int lane = tid & 31;
  const int nbatches = (N_EDGES + BATCH - 1) / BATCH;

  const uint32_t xlo = (uint32_t)(uintptr_t)x;
  const uint32_t xhi = (uint32_t)((uintptr_t)x >> 32);

  auto stage = [&](int buf, int batch) {
    const int e0 = batch * BATCH;
    if (tid < BATCH) {                       // waves 0-1: cols
      int e = e0 + tid;
      if (e < N_EDGES)
        async_ld_dword_to_lds(lds_off_of(&s_cols[buf][tid]), cols,
                              (uint32_t)e * 4u);
    } else if (tid < 2 * BATCH) {            // waves 2-3: rows
      int t = tid - BATCH, e = e0 + t;
      if (e < N_EDGES)
        async_ld_dword_to_lds(lds_off_of(&s_rows[buf][t]), rows,
                              (uint32_t)e * 4u);
    } else if (tid < 3 * BATCH) {            // waves 4-5: vals
      int t = tid - 2 * BATCH, e = e0 + t;
      if (e < N_EDGES)
        async_ld_dword_to_lds(lds_off_of(&s_vals[buf][t]), vals,
                              (uint32_t)e * 4u);
    }
  };

  // TDM gather-mode load: 8 rows of x (32-bit indices) -> per-wave LDS tile.
  auto issue_gather = [&](int tbuf, const int* cidx, int nvalid) {
    u32x4 g0;
    g0[0] = 0xC0000001u;                 // gather_mode=1, 32b idx, count=1
    g0[1] = lds_off_of(&s_xtile[tbuf][wave][0][0]);
    g0[2] = xlo;                         // global_addr[31:0]
    g0[3] = ((xhi & 0x01FFFFFFu)) | (2u << 30);  // addr[56:32] | type=2

    u32x8 g1;
    g1[0] = 2u << 16;                    // wg_mask=0, data_size=4B
    g1[1] = (uint32_t)D_FEAT << 16;      // tensor_dim0[15:0] = 128
    g1[2] = ((uint32_t)N_NODES & 0xFFFFu) << 16;          // tensor_dim1 lo
    g1[3] = ((uint32_t)N_NODES >> 16) | ((uint32_t)D_FEAT << 16); // hi|tile_dim0
    g1[4] = (uint32_t)nvalid;            // tile_dim1 = #valid indices
    g1[5] = (uint32_t)D_FEAT;            // tensor_dim0_stride = 128 elems
    g1[6] = 0u;
    g1[7] = 0u;

    u32x4 g2, g3;                        // 32-bit row indices 0..7
    g2[0] = (uint32_t)cidx[0]; g2[1] = (uint32_t)cidx[1];
    g2[2] = (uint32_t)cidx[2]; g2[3] = (uint32_t)cidx[3];
    g3[0] = (uint32_t)cidx[4]; g3[1] = (uint32_t)cidx[5];
    g3[2] = (uint32_t)cidx[6]; g3[3] = (uint32_t)cidx[7];

    asm volatile("tensor_load_to_lds %0, %1, %2, %3"
                 :: "s"(g0), "s"(g1), "s"(g2), "s"(g3)
                 : "memory");
  };

  auto scatter = [&](int tbuf, const int* pr, const float* pv, int pn) {
#pragma unroll
    for (int k = 0; k < EPW; ++k) {
      if (k >= pn) break;
      const float4 xr = *reinterpret_cast<const float4*>(
          &s_xtile[tbuf][wave][k][lane * 4]);
      const float v = pv[k];
      float* hp = h + pr[k] * D_FEAT + lane * 4;
      __hip_atomic_fetch_add(hp + 0, v * xr.x, __ATOMIC_RELAXED,
                             __HIP_MEMORY_SCOPE_AGENT);
      __hip_atomic_fetch_add(hp + 1, v * xr.y, __ATOMIC_RELAXED,
                             __HIP_MEMORY_SCOPE_AGENT);
      __hip_atomic_fetch_add(hp + 2, v * xr.z, __ATOMIC_RELAXED,
                             __HIP_MEMORY_SCOPE_AGENT);
      __hip_atomic_fetch_add(hp + 3, v * xr.w, __ATOMIC_RELAXED,
                             __HIP_MEMORY_SCOPE_AGENT);
    }
  };

  int b = blockIdx.x;
  if (b >= nbatches) return;

  stage(0, b);                            // prologue: edge batch 0 in flight
  int ebuf = 0, tbuf = 0;
  int   pr[EPW];                          // pending (previous) batch rows
  float pv[EPW];                          // pending batch vals
  int   pn = 0;

  for (; b < nbatches; b += gridDim.x) {
    wait_asynccnt0();                     // my staged edge dwords are in LDS
    __syncthreads();                      // everyone's are visible

    int nb = b + (int)gridDim.x;
    if (nb < nbatches) stage(ebuf ^ 1, nb);   // overlap next edge stage

    // Pull this wave's 8 edges; cols go uniform (SGPR) for the descriptor.
    const int e0   = b * BATCH;
    const int base = wave * EPW;
    int nvalid = N_EDGES - (e0 + base);
    nvalid = nvalid < 0 ? 0 : (nvalid > EPW ? EPW : nvalid);

    int   cidx[EPW];
    int   nr[EPW];
    float nv[EPW];
#pragma unroll
    for (int k = 0; k < EPW; ++k) {
      if (k < nvalid) {
        cidx[k] = __builtin_amdgcn_readfirstlane(s_cols[ebuf][base + k]);
        nr[k]   = s_rows[ebuf][base + k];
        nv[k]   = s_vals[ebuf][base + k];
      } else {
        cidx[k] = 0; nr[k] = 0; nv[k] = 0.f;
      }
    }
    if (nvalid > 0) issue_gather(tbuf, cidx, nvalid);  // TENSORcnt++

    // While that DMA flies, drain the previous batch's tile.
    if (pn > 0) {
      __builtin_amdgcn_s_wait_tensorcnt((short)1);  // oldest TDM retired
      scatter(tbuf ^ 1, pr, pv, pn);
    }

#pragma unroll
    for (int k = 0; k < EPW; ++k) { pr[k] = nr[k]; pv[k] = nv[k]; }
    pn = nvalid;
    tbuf ^= 1;
    ebuf ^= 1;
    __syncthreads();                      // edge buffer reusable
  }

  if (pn > 0) {                           // epilogue: last gathered tile
    __builtin_amdgcn_s_wait_tensorcnt((short)0);
    scatter(tbuf ^ 1, pr, pv, pn);
  }
}

// ---------------------------------------------------------------------------
// s = dropout(h + prev); out = (out + s) * final_scale; prev = s; h = 0
// ---------------------------------------------------------------------------
__global__ void __launch_bounds__(BLOCK)
gcn_combine(float* hbuf, float* prev, float* out,
            int layer, float final_scale, int last) {
  const int n4 = N_NODES * D_FEAT / 4;
  int i = blockIdx.x * BLOCK + threadIdx.x;
  if (i >= n4) return;

  float4 hv = reinterpret_cast<const float4*>(hbuf)[i];
  float4 pv = reinterpret_cast<const float4*>(prev)[i];
  float  s[4] = {hv.x + pv.x, hv.y + pv.y, hv.z + pv.z, hv.w + pv.w};

#pragma unroll
  for (int j = 0; j < 4; ++j) {
    float u = rng01((uint32_t)layer, (uint32_t)i * 4u + (uint32_t)j);
    s[j] = (u < KEEP) ? s[j] * (1.0f / KEEP) : 0.0f;
  }

  float4 ov = reinterpret_cast<const float4*>(out)[i];
  ov.x = (ov.x + s[0]) * final_scale;
  ov.y = (ov.y + s[1]) * final_scale;
  ov.z = (ov.z + s[2]) * final_scale;
  ov.w = (ov.w + s[3]) * final_scale;
  reinterpret_cast<float4*>(out)[i] = ov;

  if (!last) {
    reinterpret_cast<float4*>(prev)[i] = make_float4(s[0], s[1], s[2], s[3]);
    reinterpret_cast<float4*>(hbuf)[i] = make_float4(0.f, 0.f, 0.f, 0.f);
  }
}

// ---------------------------------------------------------------------------
extern "C" void kernel_launch(void* const* d_in, const int* in_sizes, int n_in,
                              void* d_out, int out_size, void* d_ws,
                              size_t ws_size, hipStream_t stream) {
  const float* x    = (const float*)d_in[0];  // question_embs [N, D]
  const int*   rows = (const int*)  d_in[1];  // edge_rows [E]
  const int*   cols = (const int*)  d_in[2];  // edge_cols [E]
  const float* vals = (const float*)d_in[3];  // edge_vals [E]
  float* out  = (float*)d_out;

  float* prev = (float*)d_ws;
  float* hbuf = prev + (size_t)N_NODES * D_FEAT;

  const int n4 = N_NODES * D_FEAT / 4;
  const int eltBlocks = (n4 + BLOCK - 1) / BLOCK;

  gcn_init<<<eltBlocks, BLOCK, 0, stream>>>(x, out, prev, hbuf);

  const int nbatches = (N_EDGES + BATCH - 1) / BATCH;
  const int spmmGrid = nbatches < 4096 ? nbatches : 4096;

  for (int layer = 0; layer < NUM_LAYERS; ++layer) {
    gcn_spmm<<<spmmGrid, BLOCK, 0, stream>>>(rows, cols, vals, prev, hbuf);
    const int   last = (layer == NUM_LAYERS - 1);
    const float fs   = last ? 1.0f / (float)(NUM_LAYERS + 1) : 1.0f;
    gcn_combine<<<eltBlocks, BLOCK, 0, stream>>>(hbuf, prev, out,
                                                 layer, fs, last);
  }
}